// DMON_13013750907216
// MI455X (gfx1250) — compile-verified
//
#include <hip/hip_runtime.h>
#include <math.h>

// Problem constants (from the reference)
constexpr int N_NODES = 16384;
constexpr int N_EDGES = 524288;
constexpr int CIN     = 128;
constexpr int HD      = 64;   // hidden == cluster count == 64
constexpr int KK      = HD * HD; // 4096

typedef float v2f __attribute__((ext_vector_type(2)));
typedef float v4f __attribute__((ext_vector_type(4)));
typedef float v8f __attribute__((ext_vector_type(8)));

// V_WMMA_F32_16X16X4_F32 : D(16x16,f32) = A(16x4,f32) * B(4x16,f32) + C
// 8-arg pattern: (neg_a, A, neg_b, B, c_mod, C, reuse_a, reuse_b)
__device__ __forceinline__ v8f wmma4(v2f a, v2f b, v8f c) {
  return __builtin_amdgcn_wmma_f32_16x16x4_f32(
      false, a, false, b, (short)0, c, false, false);
}

// ---------------------------------------------------------------- fills
__global__ void k_fill(float* p, float v, int n) {
  int i = blockIdx.x * blockDim.x + threadIdx.x;
  if (i < n) p[i] = v;
}

// ------------------------------------------------ degrees (weighted + adj)
__global__ void k_degrees(const int* __restrict__ col, const float* __restrict__ w,
                          float* __restrict__ deg, float* __restrict__ deg_adj) {
  int e = blockIdx.x * blockDim.x + threadIdx.x;
  if (e < N_EDGES) {
    int c = col[e];
    atomicAdd(&deg[c], w[e]);      // deg pre-initialized to 1.0 (self-loop weight)
    atomicAdd(&deg_adj[c], 1.0f);  // adj column counts (duplicates sum)
  }
}

__global__ void k_rsqrt_inplace(float* deg) {
  int i = blockIdx.x * blockDim.x + threadIdx.x;
  if (i < N_NODES) {
    float d = deg[i];
    deg[i] = d > 0.0f ? rsqrtf(d) : 0.0f;   // deg becomes dis
  }
}

// ------------------------------------------------ dense GEMM via f32 WMMA
// C[M,64] = A[M,Kd] (row-major) * B[Kd,64] (row-major) [+ bias]
// Block = 256 threads = 8 waves. B staged once into LDS (<=32KB), shared by
// all waves. Each wave computes a 16x64 strip: one A fragment feeds 4 WMMAs.
__global__ void k_gemm_wmma(const float* __restrict__ A, const float* __restrict__ B,
                            const float* __restrict__ bias, float* __restrict__ C,
                            int M, int Kd) {
  __shared__ float Bs[CIN * HD];                 // 32 KB (Kd <= 128)

  // cooperative B stage: Kd*64 floats, vectorized 128-bit
  {
    const v4f* Bv = (const v4f*)B;
    v4f* Sv = (v4f*)Bs;
    int nvec = (Kd * HD) >> 2;
    for (int i = threadIdx.x; i < nvec; i += blockDim.x) Sv[i] = Bv[i];
  }
  __syncthreads();

  int wave = threadIdx.x >> 5;
  int lane = threadIdx.x & 31;
  int tile = blockIdx.x * 8 + wave;              // 16-row strip index
  if (tile * 16 >= M) return;                    // wave-uniform
  int m0 = tile * 16;
  int l15 = lane & 15;
  int kb  = (lane < 16) ? 0 : 2;   // ISA A-layout: v0/v1 = K0/K1 (lanes 0-15), K2/K3 (16-31)
  int mrow = m0 + l15;

  v8f c0 = {}, c1 = {}, c2 = {}, c3 = {};
  for (int k = 0; k < Kd; k += 4) {
    v2f a;
    a.x = A[(size_t)mrow * Kd + k + kb];
    a.y = A[(size_t)mrow * Kd + k + kb + 1];
    const float* b0 = Bs + (k + kb) * HD + l15;       // row k+kb
    const float* b1 = Bs + (k + kb + 1) * HD + l15;   // row k+kb+1
    v2f b;
    b.x = b0[0];  b.y = b1[0];  c0 = wmma4(a, b, c0);
    b.x = b0[16]; b.y = b1[16]; c1 = wmma4(a, b, c1);
    b.x = b0[32]; b.y = b1[32]; c2 = wmma4(a, b, c2);
    b.x = b0[48]; b.y = b1[48]; c3 = wmma4(a, b, c3);
  }

  int rbase = m0 + ((lane >> 4) << 3);           // C layout: M = v (+8 for upper lanes)
  v8f acc[4] = {c0, c1, c2, c3};
#pragma unroll
  for (int t = 0; t < 4; ++t) {
    int colg = t * 16 + l15;
    float bv = bias ? bias[colg] : 0.0f;
#pragma unroll
    for (int v = 0; v < 8; ++v)
      C[(size_t)(rbase + v) * HD + colg] = acc[t][v] + bv;
  }
}

// ------------------------------------------------ GCN edge scatter
__global__ void k_gcn_agg(const int* __restrict__ row, const int* __restrict__ col,
                          const float* __restrict__ w, const float* __restrict__ dis,
                          const float* __restrict__ xw, float* __restrict__ agg) {
  int gid = blockIdx.x * blockDim.x + threadIdx.x;   // E*64 threads
  if (gid >= N_EDGES * HD) return;
  int e  = gid >> 6;
  int cc = gid & 63;
  int r = row[e], c = col[e];
  float norm = dis[r] * w[e] * dis[c];
  atomicAdd(&agg[c * HD + cc], norm * xw[r * HD + cc]);
}

// self-loop term + bias + relu, in place (agg becomes h)
__global__ void k_h_final(const float* __restrict__ dis, const float* __restrict__ xw,
                          const float* __restrict__ b1, float* __restrict__ agg) {
  int gid = blockIdx.x * blockDim.x + threadIdx.x;
  if (gid >= N_NODES * HD) return;
  int i = gid >> 6, cc = gid & 63;
  float d = dis[i];
  float v = agg[gid] + d * d * xw[gid] + b1[cc];
  agg[gid] = v > 0.0f ? v : 0.0f;
}

// ------------------------------------------------ row softmax (wave per row)
__global__ void k_softmax(float* __restrict__ s) {
  int wave = (blockIdx.x * blockDim.x + threadIdx.x) >> 5;
  int lane = threadIdx.x & 31;
  if (wave >= N_NODES) return;
  float* p = s + wave * HD;
  float v0 = p[lane], v1 = p[lane + 32];
  float mx = fmaxf(v0, v1);
  for (int o = 16; o > 0; o >>= 1) mx = fmaxf(mx, __shfl_xor(mx, o, 32));
  v0 = expf(v0 - mx); v1 = expf(v1 - mx);
  float sm = v0 + v1;
  for (int o = 16; o > 0; o >>= 1) sm += __shfl_xor(sm, o, 32);
  float inv = 1.0f / sm;
  p[lane] = v0 * inv; p[lane + 32] = v1 * inv;
}

// ------------------------------------------------ ss = s^T s  (WMMA, chunked)
constexpr int SS_CHUNK = 1024;
__global__ void k_ss_wmma(const float* __restrict__ s, float* __restrict__ ss) {
  int wave = (blockIdx.x * blockDim.x + threadIdx.x) >> 5;
  int lane = threadIdx.x & 31;
  const int nch = N_NODES / SS_CHUNK;     // 16
  int tile  = wave / nch;                 // 16 tiles (4x4)
  int chunk = wave % nch;
  if (tile >= 16) return;
  int ta = tile >> 2, tb = tile & 3;
  int l15 = lane & 15;
  int kb  = (lane < 16) ? 0 : 2;
  int a0 = ta * 16 + l15, b0 = tb * 16 + l15;
  int rbase = chunk * SS_CHUNK;
  v8f c = {};
  for (int r = 0; r < SS_CHUNK; r += 4) {
    const float* sr = s + (size_t)(rbase + r + kb) * HD;
    v2f a, b;
    a.x = sr[a0]; a.y = sr[HD + a0];      // A[m,k] = s[row_k, 16*ta+m]
    b.x = sr[b0]; b.y = sr[HD + b0];      // B[k,n] = s[row_k, 16*tb+n]
    c = wmma4(a, b, c);
  }
  int arow = ta * 16 + ((lane >> 4) << 3);
#pragma unroll
  for (int v = 0; v < 8; ++v)
    atomicAdd(&ss[(arow + v) * HD + tb * 16 + l15], c[v]);
}

// -------------------------- out_adj = sum_e s[row_e]^T outer s[col_e] (WMMA)
constexpr int OA_CHUNK = 2048;
__global__ void k_outadj_wmma(const float* __restrict__ s, const int* __restrict__ row,
                              const int* __restrict__ col, float* __restrict__ oa) {
  int wave = (blockIdx.x * blockDim.x + threadIdx.x) >> 5;
  int lane = threadIdx.x & 31;
  const int nch = N_EDGES / OA_CHUNK;     // 256
  int tile  = wave / nch;
  int chunk = wave % nch;
  if (tile >= 16) return;
  int ta = tile >> 2, tb = tile & 3;
  int l15 = lane & 15;
  int kb  = (lane < 16) ? 0 : 2;
  int am = ta * 16 + l15, bn = tb * 16 + l15;
  int e0 = chunk * OA_CHUNK;
  v8f c = {};
  for (int e = 0; e < OA_CHUNK; e += 4) {
    int i0 = e0 + e + kb, i1 = i0 + 1;
    if ((e & 255) == 0 && e + 256 < OA_CHUNK) {        // wave-uniform branch:
      __builtin_prefetch(&row[e0 + e + 256], 0, 0);    // global_prefetch_b8 one
      __builtin_prefetch(&col[e0 + e + 256], 0, 0);    // cacheline-run ahead
    }
    v2f a, b;
    a.x = s[(size_t)row[i0] * HD + am];   // A[m,k] = s[row_k, 16*ta+m]
    a.y = s[(size_t)row[i1] * HD + am];
    b.x = s[(size_t)col[i0] * HD + bn];   // B[k,n] = s[col_k, 16*tb+n]
    b.y = s[(size_t)col[i1] * HD + bn];
    c = wmma4(a, b, c);
  }
  int arow = ta * 16 + ((lane >> 4) << 3);
#pragma unroll
  for (int v = 0; v < 8; ++v)
    atomicAdd(&oa[(arow + v) * HD + tb * 16 + l15], c[v]);
}

// ------------------------------------------------ ca = s^T deg, csize = sum s
__global__ void k_colsums(const float* __restrict__ s, const float* __restrict__ deg_adj,
                          float* __restrict__ ca, float* __restrict__ csize) {
  int a = threadIdx.x;                    // 64 threads = 64 clusters
  int r0 = blockIdx.x * 256;
  float sca = 0.f, scs = 0.f;
  for (int r = 0; r < 256; ++r) {
    float sv = s[(size_t)(r0 + r) * HD + a];
    scs += sv;
    sca += sv * deg_adj[r0 + r];
  }
  atomicAdd(&ca[a], sca);
  atomicAdd(&csize[a], scs);
}

__global__ void k_sumdeg(const float* __restrict__ deg_adj, float* __restrict__ sum) {
  int gid = blockIdx.x * blockDim.x + threadIdx.x;
  float v = 0.f;
  for (int i = gid; i < N_NODES; i += gridDim.x * blockDim.x) v += deg_adj[i];
  for (int o = 16; o > 0; o >>= 1) v += __shfl_xor(v, o, 32);
  if ((threadIdx.x & 31) == 0) atomicAdd(sum, v);
}

// ------------------------------------------------ final scalar loss (1 block)
__global__ void k_loss(const float* __restrict__ oa, const float* __restrict__ ss,
                       const float* __restrict__ ca, const float* __restrict__ cs,
                       const float* __restrict__ sumdeg, float* __restrict__ out_loss) {
  __shared__ float red[256];
  int t = threadIdx.x;
  float a1 = 0.f;                                 // sum ss^2
  for (int i = t; i < KK; i += 256) { float v = ss[i]; a1 += v * v; }
  float a2 = 0.f, a3 = 0.f, a4 = 0.f, a5 = 0.f;
  if (t < HD) {
    a2 = ss[t * HD + t];                          // trace(ss)
    a3 = oa[t * HD + t];                          // trace(out_adj)
    a4 = ca[t] * ca[t];
    a5 = cs[t] * cs[t];
  }
  auto reduce = [&](float v) -> float {
    red[t] = v; __syncthreads();
    for (int s2 = 128; s2 > 0; s2 >>= 1) { if (t < s2) red[t] += red[t + s2]; __syncthreads(); }
    float r = red[0]; __syncthreads();
    return r;
  };
  a1 = reduce(a1); a2 = reduce(a2); a3 = reduce(a3); a4 = reduce(a4); a5 = reduce(a5);
  if (t == 0) {
    float m = sumdeg[0] * 0.5f;
    float spectral = -(a3 - a4 / (2.f * m)) / (2.f * m);
    float f = sqrtf(a1);
    // || ss/f - I/8 ||_F  with ||ss/f||_F = 1 and ||I/8||_F = 1
    float ortho = sqrtf(fmaxf(2.f - a2 / (4.f * f), 0.f));
    float cluster = sqrtf(a5) / (float)N_NODES * 8.f - 1.f;
    out_loss[0] = spectral + ortho + cluster;
  }
}

// ================================================================ launch
extern "C" void kernel_launch(void* const* d_in, const int* in_sizes, int n_in,
                              void* d_out, int out_size, void* d_ws, size_t ws_size,
                              hipStream_t stream) {
  const float* x   = (const float*)d_in[0];
  const int*   ei  = (const int*)d_in[1];           // [2, E]
  const float* ew  = (const float*)d_in[2];
  const float* W1  = (const float*)d_in[3];
  const float* b1  = (const float*)d_in[4];
  const float* Wm1 = (const float*)d_in[5];
  const float* bm1 = (const float*)d_in[6];
  const float* Wm2 = (const float*)d_in[7];
  const float* bm2 = (const float*)d_in[8];
  const int* row = ei;
  const int* col = ei + N_EDGES;

  float* out_s = (float*)d_out;                     // [N, 64]
  float* out_l = out_s + (size_t)N_NODES * HD;      // scalar loss

  const int NH = N_NODES * HD;
  float* ws      = (float*)d_ws;
  float* agg     = ws;                 // [N,64] -> becomes h
  float* xw      = agg + NH;           // [N,64] -> reused as h2
  float* deg     = xw + NH;            // [N]    -> becomes dis
  float* deg_adj = deg + N_NODES;      // [N]
  float* ss      = deg_adj + N_NODES;  // [64,64]
  float* oa      = ss + KK;            // [64,64]
  float* ca      = oa + KK;            // [64]
  float* csize   = ca + HD;            // [64]
  float* sumdeg  = csize + HD;         // [1]

  const int B = 256;
  // zero / init accumulators (deg starts at 1.0 = self-loop weight)
  k_fill<<<(NH + B - 1) / B, B, 0, stream>>>(agg, 0.f, NH);
  k_fill<<<(N_NODES + B - 1) / B, B, 0, stream>>>(deg, 1.f, N_NODES);
  int zsmall = N_NODES + KK + KK + HD + HD + 1;     // deg_adj..sumdeg contiguous
  k_fill<<<(zsmall + B - 1) / B, B, 0, stream>>>(deg_adj, 0.f, zsmall);

  k_degrees<<<(N_EDGES + B - 1) / B, B, 0, stream>>>(col, ew, deg, deg_adj);
  k_rsqrt_inplace<<<(N_NODES + B - 1) / B, B, 0, stream>>>(deg);

  // GEMM grid: 8 strips of 16 rows per block of 256 threads
  int gemm_blocks = (N_NODES / 16 + 7) / 8;         // 128

  // xw = x @ W1  (Kd = 128)
  k_gemm_wmma<<<gemm_blocks, B, 0, stream>>>(x, W1, nullptr, xw, N_NODES, CIN);

  k_gcn_agg<<<(N_EDGES * HD) / B, B, 0, stream>>>(row, col, ew, deg, xw, agg);
  k_h_final<<<(NH + B - 1) / B, B, 0, stream>>>(deg, xw, b1, agg);

  // h2 = h @ Wm1 + bm1 ; logits = h2 @ Wm2 + bm2 (into d_out)
  k_gemm_wmma<<<gemm_blocks, B, 0, stream>>>(agg, Wm1, bm1, xw, N_NODES, HD);
  k_gemm_wmma<<<gemm_blocks, B, 0, stream>>>(xw, Wm2, bm2, out_s, N_NODES, HD);

  k_softmax<<<(N_NODES * 32 + B - 1) / B, B, 0, stream>>>(out_s);

  // pooling statistics
  {
    int waves = 16 * (N_NODES / SS_CHUNK);          // 256
    k_ss_wmma<<<(waves * 32 + B - 1) / B, B, 0, stream>>>(out_s, ss);
  }
  {
    int waves = 16 * (N_EDGES / OA_CHUNK);          // 4096
    k_outadj_wmma<<<(waves * 32 + B - 1) / B, B, 0, stream>>>(out_s, row, col, oa);
  }
  k_colsums<<<N_NODES / 256, HD, 0, stream>>>(out_s, deg_adj, ca, csize);
  k_sumdeg<<<16, B, 0, stream>>>(deg_adj, sumdeg);
  k_loss<<<1, B, 0, stream>>>(oa, ss, ca, csize, sumdeg, out_l);
}